// CausalSpectralFusion_89395449299812
// MI455X (gfx1250) — compile-verified
//
#include <hip/hip_runtime.h>
#include <hip/hip_bf16.h>
#include <math.h>

// Problem constants
#define BATCH  32
#define NBANDS 5
#define CCH    64
#define TT     4096
#define DR     32          // hidden dim
#define G3     96          // 3*DR
#define PFD    4           // gi prefetch depth (async LDS ring slots)
#define STEPB  (BATCH * G3)        // floats per step block (3072)
#define STEPBY (STEPB * 4)         // bytes per step block (12288)

typedef __attribute__((ext_vector_type(2))) float v2f;
typedef __attribute__((ext_vector_type(8))) float v8f;

// Branch-free fast transcendentals (single v_exp_f32 + v_rcp_f32 each);
// the default libm paths emit IEEE-divide sequences / divergent branches,
// which sit on the 4096-step recurrence critical path.
__device__ __forceinline__ float fast_sigmoid(float x) {
  return __builtin_amdgcn_rcpf(1.f + __expf(-x));
}
__device__ __forceinline__ float fast_tanh(float x) {
  // tanh(x) = 2*sigmoid(2x) - 1
  return 2.f * __builtin_amdgcn_rcpf(1.f + __expf(-2.f * x)) - 1.f;
}

// Workgroup barrier that waits ONLY on the LDS counter (DScnt).
// __syncthreads() emits s_wait_storecnt_dscnt 0x0 / s_wait_loadcnt_dscnt 0x0,
// which drags the fire-and-forget h_hist global stores onto the serial
// recurrence critical path. LDS visibility across a workgroup barrier only
// requires DScnt==0 (CDNA5 split counters).
__device__ __forceinline__ void lds_barrier() {
  asm volatile(
      "s_wait_dscnt 0x0\n\t"
      "s_barrier_signal -1\n\t"
      "s_barrier_wait -1"
      ::: "memory");
}

// ---------------------------------------------------------------------------
// Kernel A: u = mean_c x, gi = u @ W_ih^T + b_ih, stored as (T, BATCH, 96)
// Block: (b, t-chunk); threads along t -> x reads fully coalesced.
// ---------------------------------------------------------------------------
__global__ __launch_bounds__(256) void k_gi(const float* __restrict__ x,
                                            const float* __restrict__ Wih,   // (96,5)
                                            const float* __restrict__ bih,   // (96)
                                            float* __restrict__ gi) {
  const int b = blockIdx.y;
  const int t = blockIdx.x * blockDim.x + threadIdx.x;

  float u[NBANDS];
#pragma unroll
  for (int k = 0; k < NBANDS; ++k) {
    const float* xp = x + ((size_t)(b * NBANDS + k) * CCH) * TT + t;
    float s = 0.f;
#pragma unroll
    for (int c = 0; c < CCH; ++c) s += xp[(size_t)c * TT];
    u[k] = s * (1.0f / (float)CCH);
  }

  float* gp = gi + ((size_t)t * BATCH + b) * G3;
#pragma unroll 4
  for (int j = 0; j < G3; ++j) {
    float acc = bih[j];
#pragma unroll
    for (int k = 0; k < NBANDS; ++k) acc += u[k] * Wih[j * NBANDS + k];
    gp[j] = acc;
  }
}

// ---------------------------------------------------------------------------
// Kernel B: sequential GRU scan. One workgroup, 1024 threads (32 waves).
// Per step: gh = h @ W_hh^T + b_hh via V_WMMA_F32_16X16X4_F32 (12 tiles,
// one per wave, 8 chained K=4 WMMAs each, bias folded into C init), then
// gate nonlinearities (1 thread per h element, h kept in a register).
// gi is streamed in 4 steps ahead with GLOBAL_LOAD_ASYNC_TO_LDS_B32 into a
// 4-slot LDS ring (ASYNCcnt), hiding HBM latency; each thread async-loads
// exactly the 3 values it will consume, so a wave-local s_wait_asynccnt is
// sufficient. h_t history is streamed out fire-and-forget; FC+softmax run
// in a later parallel kernel, off the critical path.
// ---------------------------------------------------------------------------
__global__ __launch_bounds__(1024) void k_scan(const float* __restrict__ gi,
                                               const float* __restrict__ Whh,  // (96,32)
                                               const float* __restrict__ bhh,  // (96)
                                               float* __restrict__ h_hist) {   // (T,32,32)
  __shared__ float h_lds[BATCH * DR];      // 4 KB, current hidden state (WMMA A frags)
  __shared__ float gh_lds[BATCH * G3];     // 12 KB, gate pre-activations
  __shared__ float gi_ring[PFD * STEPB];   // 48 KB, async-prefetched gi blocks

  const int tid  = threadIdx.x;
  const int b    = tid >> 5;    // batch row for elementwise phase
  const int j    = tid & 31;    // hidden dim for elementwise phase
  const int lane = tid & 31;    // lane within wave
  // Scalarize wave id so the WMMA region is a true scalar branch
  // (WMMA requires EXEC == all ones in the executing waves).
  const int w = __builtin_amdgcn_readfirstlane(tid >> 5);

  // --- Preload W_hh B-fragments + bias accumulator init (waves 0..11) ---
  // Tile (mt, nt): D[M=mt*16.., N=nt*16..]; B[K][N] = Whh[N][K].
  // Assumed 32-bit B 4x16 layout (mirror of documented A 16x4 layout):
  //   VGPR0: lanes0-15 K=0 / lanes16-31 K=2 ; VGPR1: K=1 / K=3, N = lane&15.
  v2f bfrag[8];
  float bias = 0.f;
  int mt = 0, nt = 0;
  if (w < 12) {
    mt = w & 1;
    nt = w >> 1;
    const int l15 = lane & 15;
    const int hi  = lane >> 4;
    const int row = nt * 16 + l15;          // N index = gate output column
#pragma unroll
    for (int c = 0; c < 8; ++c) {
      const float* wp = Whh + row * DR + c * 4 + hi * 2;
      v2f bf; bf.x = wp[0]; bf.y = wp[1];
      bfrag[c] = bf;
    }
    bias = bhh[row];                        // bias depends only on column N
  } else {
#pragma unroll
    for (int c = 0; c < 8; ++c) { v2f z; z.x = 0.f; z.y = 0.f; bfrag[c] = z; }
  }

  // --- Async gi prefetch setup ------------------------------------------
  // LDS ring slot mirrors the global per-step block layout, so ONE address
  // register pair serves both sides: the ISA adds INST_OFFSET to both the
  // LDS and the global address (offset 0/128/256 bytes = r/z/n gate rows).
  const int elem = b * G3 + j;                                   // within-block offset
  const uint64_t gsrc0   = (uint64_t)(uintptr_t)gi + (uint64_t)elem * 4;
  const uint32_t ldsbase = (uint32_t)(uintptr_t)&gi_ring[0] + (uint32_t)elem * 4;

#define ISSUE_GI_ASYNC(step, slot)                                            \
  do {                                                                        \
    uint64_t ga = gsrc0 + (uint64_t)(step) * STEPBY;                          \
    uint32_t la = ldsbase + (uint32_t)(slot) * STEPBY;                        \
    asm volatile("global_load_async_to_lds_b32 %0, %1, off offset:0"          \
                 ::"v"(la), "v"(ga) : "memory");                              \
    asm volatile("global_load_async_to_lds_b32 %0, %1, off offset:128"        \
                 ::"v"(la), "v"(ga) : "memory");                              \
    asm volatile("global_load_async_to_lds_b32 %0, %1, off offset:256"        \
                 ::"v"(la), "v"(ga) : "memory");                              \
  } while (0)

  // Prime the ring with steps 0 .. PFD-2 (slots 0 .. PFD-2).
#pragma unroll
  for (int s = 0; s < PFD - 1; ++s) ISSUE_GI_ASYNC(s, s);

  h_lds[tid] = 0.f;                         // h0 = 0
  float hcur = 0.f;                         // this thread's h[b,j], kept in register
  __syncthreads();

  for (int t = 0; t < TT; ++t) {
    // Issue async gi load for step t+PFD-1 (clamped near the end; clamped
    // duplicates rewrite identical data into the same slot -> benign).
    {
      int st = t + PFD - 1;
      if (st > TT - 1) st = TT - 1;
      ISSUE_GI_ASYNC(st, st & (PFD - 1));
    }

    // --- WMMA phase: gh = h @ W_hh^T + b_hh ---
    if (w < 12) {
      const int l15 = lane & 15;
      const int hi  = lane >> 4;
      v8f c8;
#pragma unroll
      for (int v = 0; v < 8; ++v) c8[v] = bias;
#pragma unroll
      for (int c = 0; c < 8; ++c) {
        // A 16x4 f32 layout: VGPR0 lanes0-15 K=4c+0 / lanes16-31 K=4c+2;
        // VGPR1: K=4c+1 / K=4c+3; M = lane&15 (+16*mt tile offset).
        const float* ap = &h_lds[(mt * 16 + l15) * DR + c * 4 + hi * 2];
        v2f a; a.x = ap[0]; a.y = ap[1];
        c8 = __builtin_amdgcn_wmma_f32_16x16x4_f32(
            /*neg_a=*/false, a, /*neg_b=*/false, bfrag[c],
            /*c_mod=*/(short)0, c8, /*reuse_a=*/false, /*reuse_b=*/false);
      }
      // D layout: lanes0-15 VGPR v -> M=v; lanes16-31 -> M=v+8; N = lane&15.
#pragma unroll
      for (int v = 0; v < 8; ++v) {
        const int row = mt * 16 + v + (hi ? 8 : 0);
        gh_lds[row * G3 + nt * 16 + l15] = c8[v];
      }
    }
    lds_barrier();   // waits DScnt only: gh_lds visible, async/global ops keep flying

    // Oldest async step (t) must have landed: issued steps t..t+PFD-1
    // (3 ops each) -> wait until <= 3*(PFD-1) outstanding.
    asm volatile("s_wait_asynccnt 9" ::: "memory");

    // --- Gate phase: one thread per (b, j) hidden element ---
    const float* rp  = &gi_ring[(t & (PFD - 1)) * STEPB + elem];
    const float  gr  = rp[0];
    const float  gz  = rp[32];
    const float  gn  = rp[64];
    const float ghr = gh_lds[b * G3 + j];
    const float ghz = gh_lds[b * G3 + 32 + j];
    const float ghn = gh_lds[b * G3 + 64 + j];
    const float r   = fast_sigmoid(gr + ghr);
    const float z   = fast_sigmoid(gz + ghz);
    const float n   = fast_tanh(gn + r * ghn);
    const float hn  = (1.f - z) * n + z * hcur;

    h_hist[(size_t)t * (BATCH * DR) + tid] = hn;  // fire-and-forget, never waited here
    h_lds[tid] = hn;
    hcur = hn;
    lds_barrier();   // waits DScnt only: h_lds visible for next WMMA phase,
                     // and drains ring ds_loads before slot reuse
  }
#undef ISSUE_GI_ASYNC
}

// ---------------------------------------------------------------------------
// Kernel C: alpha[b,t,:] = softmax(h_t[b] @ W_fc^T + b_fc)  (fully parallel)
// ---------------------------------------------------------------------------
__global__ __launch_bounds__(256) void k_alpha(const float* __restrict__ h_hist,
                                               const float* __restrict__ Wfc,  // (5,32)
                                               const float* __restrict__ bfc,  // (5)
                                               float* __restrict__ alpha) {    // (B,T,5)
  const int b = blockIdx.y;
  const int t = blockIdx.x * blockDim.x + threadIdx.x;

  float h[DR];
  const float4* hp = (const float4*)(h_hist + (size_t)t * (BATCH * DR) + b * DR);
#pragma unroll
  for (int i = 0; i < 8; ++i) {
    const float4 v = hp[i];
    h[4 * i + 0] = v.x; h[4 * i + 1] = v.y; h[4 * i + 2] = v.z; h[4 * i + 3] = v.w;
  }

  float logit[NBANDS];
#pragma unroll
  for (int k = 0; k < NBANDS; ++k) {
    float acc = bfc[k];
#pragma unroll
    for (int jj = 0; jj < DR; ++jj) acc += h[jj] * Wfc[k * DR + jj];
    logit[k] = acc;
  }
  float m = logit[0];
#pragma unroll
  for (int k = 1; k < NBANDS; ++k) m = fmaxf(m, logit[k]);
  float e[NBANDS], s = 0.f;
#pragma unroll
  for (int k = 0; k < NBANDS; ++k) { e[k] = __expf(logit[k] - m); s += e[k]; }
  const float inv = __builtin_amdgcn_rcpf(s);
  float* ap = alpha + ((size_t)b * TT + t) * NBANDS;
#pragma unroll
  for (int k = 0; k < NBANDS; ++k) ap[k] = e[k] * inv;
}

// ---------------------------------------------------------------------------
// Kernel D: X_fused[b,c,t] = sum_k alpha[b,t,k] * x[b,k,c,t]  (bandwidth-bound)
// ---------------------------------------------------------------------------
__global__ __launch_bounds__(256) void k_fuse(const float* __restrict__ x,
                                              const float* __restrict__ alpha,
                                              float* __restrict__ out) {
  const int b = blockIdx.z;
  const int c = blockIdx.y;
  const int t = blockIdx.x * blockDim.x + threadIdx.x;

  const float* ap = alpha + ((size_t)b * TT + t) * NBANDS;
  float s = 0.f;
#pragma unroll
  for (int k = 0; k < NBANDS; ++k)
    s += ap[k] * x[(((size_t)b * NBANDS + k) * CCH + c) * TT + t];
  out[((size_t)b * CCH + c) * TT + t] = s;
}

// ---------------------------------------------------------------------------
extern "C" void kernel_launch(void* const* d_in, const int* in_sizes, int n_in,
                              void* d_out, int out_size, void* d_ws, size_t ws_size,
                              hipStream_t stream) {
  const float* x   = (const float*)d_in[0];
  const float* Wih = (const float*)d_in[1];
  const float* Whh = (const float*)d_in[2];
  const float* bih = (const float*)d_in[3];
  const float* bhh = (const float*)d_in[4];
  const float* Wfc = (const float*)d_in[5];
  const float* bfc = (const float*)d_in[6];

  float* Xf    = (float*)d_out;                              // (32,64,4096)
  float* alpha = Xf + (size_t)BATCH * CCH * TT;              // (32,4096,5)

  float* gi     = (float*)d_ws;                              // (T,32,96)  ~50 MB
  float* h_hist = gi + (size_t)TT * BATCH * G3;              // (T,32,32)  ~17 MB

  // A: mean-over-C + input-side GRU gates
  k_gi<<<dim3(TT / 256, BATCH), 256, 0, stream>>>(x, Wih, bih, gi);
  // B: sequential GRU scan (single workgroup, fp32 WMMA per step)
  k_scan<<<1, 1024, 0, stream>>>(gi, Whh, bhh, h_hist);
  // C: FC + softmax over all timesteps (parallel)
  k_alpha<<<dim3(TT / 256, BATCH), 256, 0, stream>>>(h_hist, Wfc, bfc, alpha);
  // D: fusion einsum
  k_fuse<<<dim3(TT / 256, CCH, BATCH), 256, 0, stream>>>(x, alpha, Xf);
}